// Matchmaker_v5c_42674795053895
// MI455X (gfx1250) — compile-verified
//
#include <hip/hip_runtime.h>
#include <stdint.h>

// ---------------------------------------------------------------------------
// Conv-KNRM forward for MI455X (gfx1250, wave32).
// Heavy GEMMs (im2col convs, cosine matrices) run on V_WMMA_F32_16X16X4_F32
// (fp32 in / fp32 accumulate -> matches fp32 reference numerics).
// CDNA5 data movement: TDM tensor_load_to_lds for the qn tile (with LDS pad
// for bank-conflict-free stride), global_prefetch for the dn stream.
// ---------------------------------------------------------------------------

typedef float v2f __attribute__((ext_vector_type(2)));
typedef float v8f __attribute__((ext_vector_type(8)));
typedef unsigned int v4u __attribute__((ext_vector_type(4)));
typedef int v4i __attribute__((ext_vector_type(4)));
typedef int v8i __attribute__((ext_vector_type(8)));

// B=64, Q=32, D=512, E=300, C=256, K=11
#define BB 64
#define QQ 32
#define DD 512
#define EE 300
#define CC 256
#define KK_KER 11

__device__ __constant__ float c_mu[11]  = {1.0f, 0.9f, 0.7f, 0.5f, 0.3f, 0.1f,
                                           -0.1f, -0.3f, -0.5f, -0.7f, -0.9f};
// 1/(2*sigma^2): sigma0=0.001 -> 5e5 ; sigma=0.1 -> 50
__device__ __constant__ float c_is2[11] = {500000.0f, 50.0f, 50.0f, 50.0f, 50.0f, 50.0f,
                                           50.0f, 50.0f, 50.0f, 50.0f, 50.0f};

__device__ __forceinline__ v8f wmma_f32_4(v2f a, v2f b, v8f c) {
  // V_WMMA_F32_16X16X4_F32 : D = A(16x4) * B(4x16) + C(16x16), all fp32
  return __builtin_amdgcn_wmma_f32_16x16x4_f32(
      /*neg_a=*/false, a, /*neg_b=*/false, b,
      /*c_mod=*/(short)0, c, /*reuse_a=*/false, /*reuse_b=*/false);
}

// ---------------------------------------------------------------------------
// Kernel 0: repack conv weights into WMMA-B-friendly layout.
//   w   : [C, E, g]  ->  wtp : [KK/2][C][2]  with wtp[p][c][i] = Wt[2p+i][c],
//   Wt[kk][c] = w[c, kk%E, kk/E]  (im2col B matrix, K-major pairs adjacent).
// One-time, tiny (<= 921 KB); makes the conv B operand a single coalesced
// global_load_b64 per WMMA step.
// ---------------------------------------------------------------------------
__global__ __launch_bounds__(256) void weight_pack_kernel(
    const float* __restrict__ w, float* __restrict__ wtp, int g)
{
  const int KK = EE * g;
  const int total = KK * CC;
  const int idx = blockIdx.x * 256 + threadIdx.x;
  if (idx >= total) return;
  const int kk = idx / CC, c = idx - kk * CC;
  const int p = kk >> 1, i = kk & 1;
  const int e = kk % EE, j = kk / EE;
  wtp[((size_t)p * CC + c) * 2 + i] = w[(size_t)c * KK + e * g + j];
}

// ---------------------------------------------------------------------------
// Kernel 1: conv (kernel size g, right-pad g-1) + bias + ReLU + row L2-norm.
//   x   : [B, L, E]   (query or document embeddings, row-major)
//   wtp : packed weights (see above)
//   out : [B, L, C]   normalized rows
// Block = 256 threads (8 waves); block computes a [32 x 256] tile; each wave
// does a 2x2 grid of 16x16 WMMA tiles so one A-pair + one B-pair of b64
// loads feeds 4 WMMAs per K-step.
// ---------------------------------------------------------------------------
__global__ __launch_bounds__(256) void conv_norm_kernel(
    const float* __restrict__ x, const float* __restrict__ wtp,
    const float* __restrict__ bias, float* __restrict__ out,
    int L, int g)
{
  const int b    = blockIdx.y;
  const int row0 = blockIdx.x << 5;
  const int tid  = threadIdx.x;
  const int wave = tid >> 5;
  const int lane = tid & 31;
  const int half = lane >> 4;
  const int nn   = lane & 15;          // A row index within tile, also B col idx
  const int r0   = row0 + nn;          // M-tile 0 row
  const int r1   = r0 + 16;            // M-tile 1 row
  const float* xb = x + (size_t)b * L * EE;
  const int c0 = (wave << 5) + nn;     // N-tile 0 column (N-tile 1 = +16)

  __shared__ float yt[32 * 260];       // stride 260 -> conflict-free
  __shared__ float rs[32];

  v8f acc00 = {}; v8f acc01 = {};
  v8f acc10 = {}; v8f acc11 = {};
  for (int j = 0; j < g; ++j) {
    const bool v0 = (r0 + j) < L;      // zero right-padding (branch-free select)
    const bool v1 = (r1 + j) < L;
    const float* xr0 = xb + (size_t)(v0 ? (r0 + j) : 0) * EE;
    const float* xr1 = xb + (size_t)(v1 ? (r1 + j) : 0) * EE;
    const float* wj  = wtp + (size_t)((j * EE) >> 1) * (CC * 2);
    for (int e0 = 0; e0 < EE; e0 += 4) {   // EE % 4 == 0, never crosses j
      const int e = e0 + (half << 1);
      const float2 t0 = *(const float2*)(xr0 + e);
      const float2 t1 = *(const float2*)(xr1 + e);
      v2f a0; a0.x = v0 ? t0.x : 0.0f; a0.y = v0 ? t0.y : 0.0f;
      v2f a1; a1.x = v1 ? t1.x : 0.0f; a1.y = v1 ? t1.y : 0.0f;
      const int prow = (e0 >> 1) + half;   // K-pair index within this j block
      const float2 bq0 = *(const float2*)(wj + ((size_t)prow * CC + c0) * 2);
      const float2 bq1 = *(const float2*)(wj + ((size_t)prow * CC + c0 + 16) * 2);
      v2f b0; b0.x = bq0.x; b0.y = bq0.y;
      v2f b1; b1.x = bq1.x; b1.y = bq1.y;
      acc00 = wmma_f32_4(a0, b0, acc00);
      acc01 = wmma_f32_4(a0, b1, acc01);
      acc10 = wmma_f32_4(a1, b0, acc10);
      acc11 = wmma_f32_4(a1, b1, acc11);
    }
  }
  // bias + ReLU into LDS; lane holds rows v + 8*half (+16 for M-tile 1)
  const float bia0 = bias[c0], bia1 = bias[c0 + 16];
  for (int v = 0; v < 8; ++v) {
    const int mr = v + (half << 3);
    yt[mr * 260 + c0]             = fmaxf(acc00[v] + bia0, 0.0f);
    yt[mr * 260 + c0 + 16]        = fmaxf(acc01[v] + bia1, 0.0f);
    yt[(mr + 16) * 260 + c0]      = fmaxf(acc10[v] + bia0, 0.0f);
    yt[(mr + 16) * 260 + c0 + 16] = fmaxf(acc11[v] + bia1, 0.0f);
  }
  __syncthreads();
  if (tid < 32) {
    float s = 0.0f;
    const float* row = &yt[tid * 260];
    for (int c = 0; c < CC; ++c) { const float t = row[c]; s += t * t; }
    rs[tid] = 1.0f / (sqrtf(s) + 1e-13f);   // q / (||q|| + 1e-13)
  }
  __syncthreads();
  {
    const int mr = tid >> 3, cc = (tid & 7) << 5;
    const float sc = rs[mr];
    float* op = out + ((size_t)b * L + row0 + mr) * CC + cc;
    const float* ip = &yt[mr * 260 + cc];
    for (int c = 0; c < 32; ++c) op[c] = ip[c] * sc;
  }
}

// ---------------------------------------------------------------------------
// Kernel 2: per (pair p = qg*3+dg, batch b) cosine GEMM + all pooling.
// Block = 256 threads. qn[b] staged in LDS by the Tensor Data Mover (TDM,
// with LDS padding 4 DW per 256 DW -> stride-260 conflict-free layout);
// dn[b] streamed from L2 in 4 chunks of 128 doc columns; cos chunk parked
// in LDS for the Gaussian-kernel pooling.
// ---------------------------------------------------------------------------
__global__ __launch_bounds__(256) void pair_pool_kernel(
    const float* __restrict__ qn_all, const float* __restrict__ dn_all,
    const float* __restrict__ qmask,  const float* __restrict__ dmask,
    const float* __restrict__ scaler_p,
    float* __restrict__ per_kernel, float* __restrict__ per_mean,
    float* __restrict__ prox)
{
  const int p = blockIdx.x, b = blockIdx.y;
  const int qg = p / 3, dg = p % 3;
  const float* qb  = qn_all + ((size_t)qg * BB + b) * (QQ * CC);
  const float* dbp = dn_all + ((size_t)dg * BB + b) * (DD * CC);
  const int tid  = threadIdx.x;
  const int wave = tid >> 5;
  const int lane = tid & 31;
  const int half = lane >> 4;
  const int nn   = lane & 15;

  __shared__ float qt[QQ * 260];      // qn tile, stride 260 (TDM-padded)
  __shared__ float cosL[QQ * 132];    // 32 x 128 cos chunk, stride 132
  __shared__ float pkqL[QQ * 12];     // pkq accumulators [q][k]
  __shared__ float dmL[DD];
  __shared__ float qmL[QQ];
  __shared__ float mpdf[DD];          // max-per-doc-position
  __shared__ float csum[DD + 1];
  __shared__ float part[13 * 8 * 11]; // proximity partial sums
  __shared__ float s_doclen;

#if __has_builtin(__builtin_amdgcn_tensor_load_to_lds)
  // ---- TDM: DMA the 32x256 fp32 qn tile into LDS, padding +4 DWORDs per
  //      256 DWORDs so the LDS image lands at stride 260. Wave 0 issues.
  if (tid < 32) {
    const uint64_t ga = (uint64_t)(uintptr_t)qb;
    const unsigned lds_off = (unsigned)(uintptr_t)&qt[0]; // low 32b = LDS offset
    v4u g0;
    g0[0] = 1u;                                   // count=1 (valid descriptor)
    g0[1] = lds_off;                              // lds_addr
    g0[2] = (unsigned)(ga & 0xFFFFFFFFu);         // global_addr[31:0]
    g0[3] = (unsigned)((ga >> 32) & 0x01FFFFFFu)  // global_addr[56:32]
            | (2u << 30);                         // type = 2 ("image")
    v8i g1;
    g1[0] = (int)((2u << 16)      // data_size = 4B
                | (1u << 20)      // pad_enable
                | (7u << 22)      // pad_interval = 256 DWORDs
                | (3u << 25));    // pad_amount  = 4 DWORDs
    g1[1] = (int)(256u << 16);    // tensor_dim0 = 256 (low 16 bits)
    g1[2] = (int)(32u << 16);     // tensor_dim0 hi=0 | tensor_dim1 = 32
    g1[3] = (int)(256u << 16);    // tensor_dim1 hi=0 | tile_dim0 = 256
    g1[4] = 32;                   // tile_dim1 = 32, tile_dim2 = 0
    g1[5] = 256;                  // tensor_dim0_stride = 256
    g1[6] = 0;
    g1[7] = 0;
    const v4i gz = {0, 0, 0, 0};
#if defined(__clang_major__) && (__clang_major__ >= 23)
    const v8i gz8 = {0, 0, 0, 0, 0, 0, 0, 0};
    __builtin_amdgcn_tensor_load_to_lds(g0, g1, gz, gz, gz8, 0);
#else
    __builtin_amdgcn_tensor_load_to_lds(g0, g1, gz, gz, 0);
#endif
    __builtin_amdgcn_s_wait_tensorcnt(0);
  }
#else
  for (int i = tid; i < QQ * CC; i += 256) qt[(i >> 8) * 260 + (i & 255)] = qb[i];
#endif
  for (int i = tid; i < DD; i += 256) dmL[i] = dmask[(size_t)b * DD + i];
  if (tid < QQ) qmL[tid] = qmask[(size_t)b * QQ + tid];
  for (int i = tid; i < QQ * 12; i += 256) pkqL[i] = 0.0f;
  __syncthreads();

  const float scl = scaler_p[0];

  for (int chunk = 0; chunk < 4; ++chunk) {
    // ---- cosine GEMM: wave handles 16 doc columns, both 16-row halves ----
    const int dcol = (chunk << 7) + (wave << 4) + nn;   // global doc column
    const int lcol = (wave << 4) + nn;                  // local column
    const float* dpc = dbp + (size_t)dcol * CC;
    if (chunk < 3) __builtin_prefetch(dpc + 128 * CC, 0, 1);  // warm next chunk
    v8f acc0 = {};
    v8f acc1 = {};
    const float* qr0 = &qt[nn * 260];
    const float* qr1 = &qt[(nn + 16) * 260];
    for (int s = 0; s < 64; ++s) {           // K = 256 = 64 steps of 4
      const int kb = (s << 2) + (half << 1);
      const float2 bq = *(const float2*)(dpc + kb);
      const float2 a0 = *(const float2*)(qr0 + kb);
      const float2 a1 = *(const float2*)(qr1 + kb);
      v2f bb;  bb.x  = bq.x; bb.y  = bq.y;
      v2f av0; av0.x = a0.x; av0.y = a0.y;
      v2f av1; av1.x = a1.x; av1.y = a1.y;
      acc0 = wmma_f32_4(av0, bb, acc0);      // rows 0..15
      acc1 = wmma_f32_4(av1, bb, acc1);      // rows 16..31
    }
    const float dmv = dmL[dcol];
    for (int v = 0; v < 8; ++v) {
      const int mr = v + (half << 3);
      cosL[mr * 132 + lcol]        = acc0[v] * qmL[mr]      * dmv;  // cos * qd_mask
      cosL[(mr + 16) * 132 + lcol] = acc1[v] * qmL[mr + 16] * dmv;
    }
    __syncthreads();

    // ---- mpd partial: max over q for this chunk's 128 doc positions ----
    if (tid < 128) {
      float m = -3.0e38f;
      for (int q = 0; q < QQ; ++q) m = fmaxf(m, cosL[q * 132 + tid]);
      mpdf[(chunk << 7) + tid] = m;
    }
    // ---- pkq partial: sum_d exp(-(cos-mu)^2/(2s^2)) * qd_mask ----
    for (int idx = tid; idx < QQ * KK_KER; idx += 256) {
      const int q = idx / KK_KER, k = idx - q * KK_KER;
      const float mu = c_mu[k], is2 = c_is2[k];
      const float* cr  = &cosL[q * 132];
      const float* dmc = &dmL[chunk << 7];
      float s = 0.0f;
      for (int dl = 0; dl < 128; ++dl) {
        const float t = cr[dl] - mu;
        s += __expf(-t * t * is2) * dmc[dl];
      }
      pkqL[q * 12 + k] += qmL[q] * s;
    }
    __syncthreads();
  }

  // ---- doc_len and cumsum of mpd (cheap serial work) ----
  if (tid == 0) {
    float s = 0.0f;
    for (int i = 0; i < DD; ++i) s += dmL[i];
    s_doclen = s;
    float c = 0.0f;
    csum[0] = 0.0f;
    for (int i = 0; i < DD; ++i) { c += mpdf[i]; csum[i + 1] = c; }
  }
  __syncthreads();

  // ---- per_kernel / per_kernel_mean ----
  if (tid >= 32 && tid < 32 + KK_KER) {
    const int k = tid - 32;
    const float inv_dl = 1.0f / s_doclen;
    float s1 = 0.0f, s2 = 0.0f;
    for (int q = 0; q < QQ; ++q) {
      const float pk = pkqL[q * 12 + k];
      const float qm = qmL[q];
      s1 += __logf(fmaxf(pk, 1e-10f)) * scl * qm;
      s2 += __logf(fmaxf(pk * inv_dl, 1e-10f)) * scl * qm;
    }
    const size_t o = ((size_t)p * BB + b) * KK_KER + k;
    per_kernel[o] = s1;
    per_mean[o]   = s2;
  }
  // ---- proximity: sliding window means (w=2..14) -> Gaussian -> sum_d ----
  if (tid >= 64 && tid < 64 + 13 * 8) {
    const int t = tid - 64;
    const int wi = t >> 3, sub = t & 7;
    const int wv = wi + 2;
    const float invw = 1.0f / (float)wv;
    float acc[KK_KER];
    for (int k = 0; k < KK_KER; ++k) acc[k] = 0.0f;
    for (int d = sub; d < DD; d += 8) {
      // padded tail positions contribute value 0.0 through the kernel (ref pads)
      const float val = (d + wv <= DD) ? (csum[d + wv] - csum[d]) * invw : 0.0f;
      const float dmv = dmL[d];
      for (int k = 0; k < KK_KER; ++k) {
        const float t2 = val - c_mu[k];
        acc[k] += __expf(-t2 * t2 * c_is2[k]) * dmv;
      }
    }
    for (int k = 0; k < KK_KER; ++k) part[(wi * 8 + sub) * KK_KER + k] = acc[k];
  }
  __syncthreads();
  if (tid < 13 * KK_KER) {
    const int wi = tid / KK_KER, k = tid - wi * KK_KER;
    float s = 0.0f;
    for (int sub = 0; sub < 8; ++sub) s += part[(wi * 8 + sub) * KK_KER + k];
    prox[(((size_t)p * BB + b) * 13 + wi) * KK_KER + k] = s * scl;
  }
}

// ---------------------------------------------------------------------------
// Kernel 3: tiny dense heads + final combination. One thread per batch row.
// ---------------------------------------------------------------------------
__global__ __launch_bounds__(64) void finalize_kernel(
    const float* __restrict__ perk, const float* __restrict__ perm,
    const float* __restrict__ prox,
    const float* __restrict__ pw,  const float* __restrict__ pb,
    const float* __restrict__ pcw, const float* __restrict__ pcb,
    const float* __restrict__ dw,  const float* __restrict__ dbv,
    const float* __restrict__ dmw, const float* __restrict__ dmb,
    const float* __restrict__ cbw, float* __restrict__ out)
{
  const int b = threadIdx.x;
  if (b >= BB) return;
  float dout = dbv[0], dmean = dmb[0];
  for (int p = 0; p < 9; ++p)
    for (int k = 0; k < KK_KER; ++k) {
      const int f = p * KK_KER + k;
      const size_t o = ((size_t)p * BB + b) * KK_KER + k;
      dout  += perk[o] * dw[f];
      dmean += perm[o] * dmw[f];
    }
  float pos_lin = pcb[0];
  for (int pp = 0; pp < 13; ++pp) {
    float s = pb[pp];
    for (int p = 0; p < 9; ++p)
      for (int k = 0; k < KK_KER; ++k) {
        const int f = p * KK_KER + k;
        s += prox[(((size_t)p * BB + b) * 13 + pp) * KK_KER + k] * pw[pp * 99 + f];
      }
    pos_lin += s * pcw[pp];
  }
  out[b] = cbw[0] * dout + cbw[1] * dmean + cbw[2] * pos_lin;
}

// ---------------------------------------------------------------------------
extern "C" void kernel_launch(void* const* d_in, const int* in_sizes, int n_in,
                              void* d_out, int out_size, void* d_ws, size_t ws_size,
                              hipStream_t stream)
{
  const float* qe  = (const float*)d_in[0];
  const float* de  = (const float*)d_in[1];
  const float* qm  = (const float*)d_in[2];
  const float* dm  = (const float*)d_in[3];
  const float* cw[3] = {(const float*)d_in[4], (const float*)d_in[6], (const float*)d_in[8]};
  const float* cb[3] = {(const float*)d_in[5], (const float*)d_in[7], (const float*)d_in[9]};
  const float* sc  = (const float*)d_in[10];
  const float* pw  = (const float*)d_in[11];
  const float* pb  = (const float*)d_in[12];
  const float* pcw = (const float*)d_in[13];
  const float* pcb = (const float*)d_in[14];
  const float* dw  = (const float*)d_in[15];
  const float* dbv = (const float*)d_in[16];
  const float* dmw = (const float*)d_in[17];
  const float* dmb = (const float*)d_in[18];
  const float* cbw = (const float*)d_in[19];
  float* outp = (float*)d_out;

  // Workspace (floats): qn[3][B][Q][C] | dn[3][B][D][C] | wtp | perk | perm | prox
  float* ws = (float*)d_ws;
  const size_t QN_SZ = 3ull * BB * QQ * CC;        //  1.5 M floats
  const size_t DN_SZ = 3ull * BB * DD * CC;        // 25.2 M floats
  const size_t WT_SZ = (300ull + 600ull + 900ull) * CC;  // packed weights
  float* qn   = ws;
  float* dn   = qn + QN_SZ;
  float* wtp  = dn + DN_SZ;
  float* perk = wtp + WT_SZ;
  float* perm = perk + 9ull * BB * KK_KER;
  float* prox = perm + 9ull * BB * KK_KER;
  const size_t wt_off[3] = {0, 300ull * CC, 900ull * CC};

  for (int g = 1; g <= 3; ++g) {
    const int total = EE * g * CC;
    weight_pack_kernel<<<(total + 255) / 256, 256, 0, stream>>>(
        cw[g - 1], wtp + wt_off[g - 1], g);
  }
  for (int g = 1; g <= 3; ++g) {
    conv_norm_kernel<<<dim3(QQ / 32, BB), 256, 0, stream>>>(
        qe, wtp + wt_off[g - 1], cb[g - 1],
        qn + (size_t)(g - 1) * BB * QQ * CC, QQ, g);
    conv_norm_kernel<<<dim3(DD / 32, BB), 256, 0, stream>>>(
        de, wtp + wt_off[g - 1], cb[g - 1],
        dn + (size_t)(g - 1) * BB * DD * CC, DD, g);
  }
  pair_pool_kernel<<<dim3(9, BB), 256, 0, stream>>>(qn, dn, qm, dm, sc, perk, perm, prox);
  finalize_kernel<<<1, 64, 0, stream>>>(perk, perm, prox, pw, pb, pcw, pcb,
                                        dw, dbv, dmw, dmb, cbw, outp);
  (void)in_sizes; (void)n_in; (void)out_size; (void)ws_size;
}